// EllipseProposalLayer_41901700939961
// MI455X (gfx1250) — compile-verified
//
#include <hip/hip_runtime.h>
#include <math.h>

// ---------------------------------------------------------------------------
// EllipseProposalLayer for MI455X (gfx1250, wave32).
// Pipeline: decode+score -> bitonic top-6000 -> WMMA-assisted IoU bitmask ->
// single-wave sequential NMS (barriers are S_NOP for 1-wave WGs) -> compact.
// IoU test done division-free: inter > th*(union), union > 0 always.
// ---------------------------------------------------------------------------

#define NANCH   81920      // 128*128*5
#define MSORT   131072     // next pow2
#define TOPK    6000
#define POSTK   300
#define NWORDS  188        // ceil(6000/32)
#define NMS_TH  0.7f
#define PADF    4.0f
#define IMGMAX  1023.0f
#define MINSZ   8.0f

typedef __attribute__((ext_vector_type(2))) float v2f;
typedef __attribute__((ext_vector_type(8))) float v8f;
typedef unsigned long long u64;
typedef unsigned int u32;

// ---- workspace layout (bytes) ----
#define ELL_OFF  0u                     // N*5*4   = 1,638,400
#define BOX_OFF  1638400u               // N*4*4   = 1,310,720
#define SC_OFF   2949120u               // N*4     =   327,680
#define KEY_OFF  3276800u               // M*8     = 1,048,576
#define BT_OFF   4325376u               // 6000*4*4=    96,000
#define ET_OFF   4421376u               // 6000*5*4=   120,000
#define ST_OFF   4541376u               // 6000*4  =    24,000
#define AR_OFF   4565376u               // 6000*4  =    24,000
#define MK_OFF   4589376u               // 6000*188*4 = 4,512,000  (end ~9.1MB)

// ===========================================================================
// 1) decode: softmax score, ellipse decode, box, clip, validity, sort keys
// ===========================================================================
__global__ void decode_kernel(const float* __restrict__ cls,
                              const float* __restrict__ del,
                              const float* __restrict__ anch,
                              float* __restrict__ ell,
                              float* __restrict__ boxes,
                              float* __restrict__ sc,
                              u64* __restrict__ keys) {
  int idx = blockIdx.x * blockDim.x + threadIdx.x;
  if (idx >= MSORT) return;
  if (idx >= NANCH) { keys[idx] = 0ull; return; }   // pad sorts last

  float c0 = cls[2 * idx + 0];
  float c1 = cls[2 * idx + 1];
  float score = 1.0f / (1.0f + expf(c0 - c1));       // softmax[...,1]

  float a0 = anch[4 * idx + 0], a1 = anch[4 * idx + 1];
  float a2 = anch[4 * idx + 2], a3 = anch[4 * idx + 3];
  float w = a2 - a0 + 1.0f, h = a3 - a1 + 1.0f;
  float cx = a0 + 0.5f * w,  cy = a1 + 0.5f * h;

  float d0 = del[5 * idx + 0], d1 = del[5 * idx + 1];
  float d2 = del[5 * idx + 2], d3 = del[5 * idx + 3];
  float d4 = del[5 * idx + 4];

  float px = d0 * w + cx;
  float py = d1 * h + cy;
  float pa = expf(d2) * w * 0.5f;
  float pb = expf(d3) * h * 0.5f;
  float pt = d4;

  float co = cosf(pt), si = sinf(pt);
  float ux = sqrtf(pa * co * pa * co + pb * si * pb * si);
  float uy = sqrtf(pa * si * pa * si + pb * co * pb * co);

  float x1 = fminf(fmaxf(px - ux - PADF, 0.0f), IMGMAX);
  float y1 = fminf(fmaxf(py - uy - PADF, 0.0f), IMGMAX);
  float x2 = fminf(fmaxf(px + ux + PADF, 0.0f), IMGMAX);
  float y2 = fminf(fmaxf(py + uy + PADF, 0.0f), IMGMAX);

  float bw = x2 - x1 + 1.0f, bh = y2 - y1 + 1.0f;
  bool valid = (bw >= MINSZ) && (bh >= MINSZ);
  float sv = valid ? score : -INFINITY;

  ell[5 * idx + 0] = px; ell[5 * idx + 1] = py; ell[5 * idx + 2] = pa;
  ell[5 * idx + 3] = pb; ell[5 * idx + 4] = pt;
  boxes[4 * idx + 0] = x1; boxes[4 * idx + 1] = y1;
  boxes[4 * idx + 2] = x2; boxes[4 * idx + 3] = y2;
  sc[idx] = sv;

  // order-preserving float->uint map; descending (score, then ascending idx)
  u32 u = __float_as_uint(sv);
  u = (u & 0x80000000u) ? ~u : (u | 0x80000000u);
  keys[idx] = ((u64)u << 32) | (u32)(~(u32)idx);
}

// ===========================================================================
// 2) bitonic sort (descending) on u64 keys, LDS-accelerated for spans<=512
// ===========================================================================
__device__ inline void cmpxchg_desc(u64& a, u64& b, bool dirDesc) {
  bool sw = dirDesc ? (a < b) : (a > b);
  if (sw) { u64 t = a; a = b; b = t; }
}

__global__ void bitonic_local_full(u64* __restrict__ keys) {
  __shared__ u64 sh[1024];
  int base = blockIdx.x * 1024;
  int t = threadIdx.x;                       // 0..511
  sh[t] = keys[base + t];
  sh[t + 512] = keys[base + t + 512];
  __syncthreads();
  for (int k = 2; k <= 1024; k <<= 1) {
    for (int j = k >> 1; j >= 1; j >>= 1) {
      int i = 2 * j * (t / j) + (t % j);
      bool dirDesc = (((base + i) & k) == 0);
      cmpxchg_desc(sh[i], sh[i + j], dirDesc);
      __syncthreads();
    }
  }
  keys[base + t] = sh[t];
  keys[base + t + 512] = sh[t + 512];
}

__global__ void bitonic_global(u64* __restrict__ keys, int k, int j) {
  int t = blockIdx.x * blockDim.x + threadIdx.x;   // 0..MSORT/2-1
  int i = 2 * j * (t / j) + (t % j);
  bool dirDesc = ((i & k) == 0);
  u64 a = keys[i], b = keys[i + j];
  bool sw = dirDesc ? (a < b) : (a > b);
  if (sw) { keys[i] = b; keys[i + j] = a; }
}

__global__ void bitonic_local_finish(u64* __restrict__ keys, int k) {
  __shared__ u64 sh[1024];
  int base = blockIdx.x * 1024;
  int t = threadIdx.x;
  sh[t] = keys[base + t];
  sh[t + 512] = keys[base + t + 512];
  __syncthreads();
  bool dirDesc = ((base & k) == 0);          // uniform per block (k > 1024)
  for (int j = 512; j >= 1; j >>= 1) {
    int i = 2 * j * (t / j) + (t % j);
    cmpxchg_desc(sh[i], sh[i + j], dirDesc);
    __syncthreads();
  }
  keys[base + t] = sh[t];
  keys[base + t + 512] = sh[t + 512];
}

// ===========================================================================
// 3) gather top-6000 into compact arrays + areas
// ===========================================================================
__global__ void gather_topk(const u64* __restrict__ keys,
                            const float* __restrict__ boxes,
                            const float* __restrict__ ell,
                            const float* __restrict__ sc,
                            float* __restrict__ bt, float* __restrict__ et,
                            float* __restrict__ st, float* __restrict__ areas) {
  int t = blockIdx.x * blockDim.x + threadIdx.x;
  if (t >= TOPK) return;
  int idx = (int)(~(u32)keys[t]);
  if (idx < 0 || idx >= NANCH) idx = 0;      // padding safety (can't reach topk)
  float x1 = boxes[4 * idx + 0], y1 = boxes[4 * idx + 1];
  float x2 = boxes[4 * idx + 2], y2 = boxes[4 * idx + 3];
  bt[4 * t + 0] = x1; bt[4 * t + 1] = y1; bt[4 * t + 2] = x2; bt[4 * t + 3] = y2;
  for (int q = 0; q < 5; ++q) et[5 * t + q] = ell[5 * idx + q];
  st[t] = sc[idx];
  areas[t] = (x2 - x1 + 1.0f) * (y2 - y1 + 1.0f);
}

// ===========================================================================
// 4) suppression bitmask: 1 wave per 16x32 tile; WMMA computes area[i]+area[j]
//    division-free IoU test: inter > th * (areaSum - inter); union > 0 always.
// ===========================================================================
__device__ inline bool iou_gt(const float* __restrict__ A,
                              const float* __restrict__ B,
                              float areaSum, int i, int j) {
  float xx1 = fmaxf(A[0], B[0]);
  float yy1 = fmaxf(A[1], B[1]);
  float xx2 = fminf(A[2], B[2]);
  float yy2 = fminf(A[3], B[3]);
  float iw = fmaxf(xx2 - xx1 + 1.0f, 0.0f);
  float ih = fmaxf(yy2 - yy1 + 1.0f, 0.0f);
  float inter = iw * ih;
  // iou > th  <=>  inter > th * union   (union = areaSum - inter > 0)
  bool hit = inter > NMS_TH * (areaSum - inter);
  return hit && (j > i) && (j < TOPK);
}

__global__ __launch_bounds__(32) void mask_kernel(const float* __restrict__ bt,
                                                  const float* __restrict__ areas,
                                                  u32* __restrict__ mask) {
  __shared__ float rb[64];    // 16 row boxes
  __shared__ float cb[128];   // 32 col boxes
  __shared__ float ra[16];
  __shared__ float ca[32];
  int lane = threadIdx.x;
  int rowBase = blockIdx.x * 16;
  int colWord = blockIdx.y;
  int colBase = colWord * 32;

  if (lane < 16) ra[lane] = areas[rowBase + lane];
  { int ci = colBase + lane; ca[lane] = (ci < TOPK) ? areas[ci] : 1.0f; }
  for (int r = lane; r < 64; r += 32) rb[r] = bt[rowBase * 4 + r];
  for (int r = lane; r < 128; r += 32) {
    int gi = colBase * 4 + r;
    cb[r] = (gi < TOPK * 4) ? bt[gi] : 0.0f;
  }
  __syncthreads();

  // Outer-sum D[m][n] = area_i[m] + area_j[n] via A(16x4)[area,1,0,0] x
  // B(4x16)[1;area;0;0] on the matrix unit (wave32, EXEC all ones).
  v2f a  = {0.0f, 0.0f};
  v2f b0 = {0.0f, 0.0f};
  v2f b1 = {0.0f, 0.0f};
  if (lane < 16) {
    a  = (v2f){ra[lane], 1.0f};
    b0 = (v2f){1.0f, ca[lane]};
    b1 = (v2f){1.0f, ca[lane + 16]};
  }
  v8f cz = {};
  v8f d0, d1;
#if __has_builtin(__builtin_amdgcn_wmma_f32_16x16x4_f32)
  d0 = __builtin_amdgcn_wmma_f32_16x16x4_f32(false, a, false, b0, (short)0, cz,
                                             false, false);
  d1 = __builtin_amdgcn_wmma_f32_16x16x4_f32(false, a, false, b1, (short)0, cz,
                                             false, false);
#else
  #pragma unroll
  for (int r = 0; r < 8; ++r) {
    int m = r + ((lane >> 4) << 3);
    int n = lane & 15;
    d0[r] = ra[m] + ca[n];
    d1[r] = ra[m] + ca[n + 16];
  }
#endif

  #pragma unroll
  for (int r = 0; r < 8; ++r) {
    // D layout: lane L, VGPR r -> (M = r + 8*(L>=16), N = L&15)
    int m = r + ((lane >> 4) << 3);
    int n0 = lane & 15;
    int gi = rowBase + m;
    bool p0 = iou_gt(&rb[m * 4], &cb[n0 * 4],        d0[r], gi, colBase + n0);
    bool p1 = iou_gt(&rb[m * 4], &cb[(n0 + 16) * 4], d1[r], gi, colBase + n0 + 16);
    u32 m0 = (u32)__ballot(p0 ? 1 : 0);
    u32 m1 = (u32)__ballot(p1 ? 1 : 0);
    // lanes 0-15 of m0/m1 are row r (cols 0-15 / 16-31); lanes 16-31 row r+8
    u32 wlo = (m0 & 0xFFFFu) | ((m1 & 0xFFFFu) << 16);
    u32 whi = (m0 >> 16) | (m1 & 0xFFFF0000u);
    if (lane == 0) {
      mask[(size_t)(rowBase + r) * NWORDS + colWord] = wlo;
      mask[(size_t)(rowBase + r + 8) * NWORDS + colWord] = whi;
    }
  }
}

// ===========================================================================
// 5) sequential greedy NMS on one wave (workgroup barriers are S_NOP) + output
// ===========================================================================
__global__ __launch_bounds__(32) void nms_out_kernel(const u32* __restrict__ mask,
                                                     const float* __restrict__ bt,
                                                     const float* __restrict__ et,
                                                     const float* __restrict__ st,
                                                     float* __restrict__ out) {
  __shared__ u32 rem[NWORDS];
  __shared__ int kept[POSTK];
  __shared__ int keptCnt;
  int lane = threadIdx.x;

  // initial keep = isfinite(score); removed bit set where invalid or t>=TOPK
  for (int w = lane; w < NWORDS; w += 32) {
    u32 v = 0;
    for (int b = 0; b < 32; ++b) {
      int t = w * 32 + b;
      if (t >= TOPK || !(st[t] > -1e37f)) v |= (1u << b);
    }
    rem[w] = v;
  }
  __syncthreads();

  for (int i = 0; i < TOPK; ++i) {
    if (i + 1 < TOPK && lane < 6) {
      // hide L2 latency of the dependent chain: global_prefetch_b8 next row
      __builtin_prefetch(&mask[(size_t)(i + 1) * NWORDS + lane * 32], 0, 0);
    }
    __syncthreads();
    u32 rw = rem[i >> 5];
    if (!((rw >> (i & 31)) & 1u)) {
      for (int w = lane; w < NWORDS; w += 32)
        rem[w] |= mask[(size_t)i * NWORDS + w];
    }
  }
  __syncthreads();

  if (lane == 0) {
    int c = 0;
    for (int t = 0; t < TOPK && c < POSTK; ++t)
      if (!((rem[t >> 5] >> (t & 31)) & 1u)) kept[c++] = t;
    keptCnt = c;
  }
  __syncthreads();

  int kc = keptCnt;
  // out: boxes [0,1200) | ell [1200,2700) | scores [2700,3000)
  for (int s = lane; s < POSTK; s += 32) {
    bool v = (s < kc);
    int t = v ? kept[s] : 0;
    out[4 * s + 0] = v ? bt[4 * t + 0] : 0.0f;
    out[4 * s + 1] = v ? bt[4 * t + 1] : 0.0f;
    out[4 * s + 2] = v ? bt[4 * t + 2] : 0.0f;
    out[4 * s + 3] = v ? bt[4 * t + 3] : 0.0f;
    for (int q = 0; q < 5; ++q)
      out[1200 + 5 * s + q] = v ? et[5 * t + q] : 0.0f;
    out[2700 + s] = v ? st[t] : 0.0f;
  }
}

// ===========================================================================
extern "C" void kernel_launch(void* const* d_in, const int* in_sizes, int n_in,
                              void* d_out, int out_size, void* d_ws, size_t ws_size,
                              hipStream_t stream) {
  const float* cls  = (const float*)d_in[0];
  const float* del  = (const float*)d_in[1];
  const float* anch = (const float*)d_in[2];
  char* ws = (char*)d_ws;

  float* ell   = (float*)(ws + ELL_OFF);
  float* boxes = (float*)(ws + BOX_OFF);
  float* sc    = (float*)(ws + SC_OFF);
  u64*   keys  = (u64*)(ws + KEY_OFF);
  float* bt    = (float*)(ws + BT_OFF);
  float* et    = (float*)(ws + ET_OFF);
  float* st    = (float*)(ws + ST_OFF);
  float* areas = (float*)(ws + AR_OFF);
  u32*   mask  = (u32*)(ws + MK_OFF);

  decode_kernel<<<MSORT / 256, 256, 0, stream>>>(cls, del, anch, ell, boxes, sc, keys);

  bitonic_local_full<<<MSORT / 1024, 512, 0, stream>>>(keys);
  for (int k = 2048; k <= MSORT; k <<= 1) {
    for (int j = k >> 1; j >= 1024; j >>= 1)
      bitonic_global<<<(MSORT / 2) / 256, 256, 0, stream>>>(keys, k, j);
    bitonic_local_finish<<<MSORT / 1024, 512, 0, stream>>>(keys, k);
  }

  gather_topk<<<(TOPK + 255) / 256, 256, 0, stream>>>(keys, boxes, ell, sc,
                                                      bt, et, st, areas);

  dim3 mg(TOPK / 16, NWORDS);
  mask_kernel<<<mg, 32, 0, stream>>>(bt, areas, mask);

  nms_out_kernel<<<1, 32, 0, stream>>>(mask, bt, et, st, (float*)d_out);
}